// SCM_block_87789131530335
// MI455X (gfx1250) — compile-verified
//
#include <hip/hip_runtime.h>

// ---------------------------------------------------------------------------
// MI455X (gfx1250) implementation: wave32, WMMA f32 16x16x4 matrix core path.
// ---------------------------------------------------------------------------

#define Bn   2
#define Cn   64
#define Hf   512
#define Wf   512
#define Hp   256
#define Wp   256
#define HWF  (Hf*Wf)     // 262144
#define HWP  (Hp*Wp)     // 65536

typedef float v2f __attribute__((ext_vector_type(2)));
typedef float v8f __attribute__((ext_vector_type(8)));

// D = A(16x4) * B(4x16) + C, fp32 matrix pipe.
__device__ __forceinline__ v8f wmma4(v2f a, v2f b, v8f c) {
  return __builtin_amdgcn_wmma_f32_16x16x4_f32(false, a, false, b, (short)0, c, false, false);
}

// ---------------------------------------------------------------------------
// maxpool 4x4 stride 2 pad 1 (-inf identity): [B,64,512,512] -> [B,64,256,256]
// ---------------------------------------------------------------------------
__global__ __launch_bounds__(256) void maxpool_kernel(const float* __restrict__ x,
                                                      float* __restrict__ y) {
  int idx = blockIdx.x * 256 + threadIdx.x;            // Bn*Cn*HWP threads
  int ox = idx & 255, oy = (idx >> 8) & 255, bc = idx >> 16;
  const float* xin = x + (size_t)bc * HWF;
  int iy0 = 2 * oy - 1, ix0 = 2 * ox - 1;
  float m = -INFINITY;
  #pragma unroll
  for (int dy = 0; dy < 4; ++dy) {
    int iy = iy0 + dy;
    if ((unsigned)iy < (unsigned)Hf) {
      const float* row = xin + (size_t)iy * Wf;
      #pragma unroll
      for (int dx = 0; dx < 4; ++dx) {
        int ix = ix0 + dx;
        if ((unsigned)ix < (unsigned)Wf) m = fmaxf(m, row[ix]);
      }
    }
  }
  y[idx] = m;
}

// ---------------------------------------------------------------------------
// zero GN stats (sum[32], sumsq[32])
// ---------------------------------------------------------------------------
__global__ void zero_stats_kernel(float* stats) {
  if (threadIdx.x < 64) stats[threadIdx.x] = 0.0f;
}

// ---------------------------------------------------------------------------
// conv1x1 (y = W[64x64] @ x + bias) via WMMA + GroupNorm stat accumulation.
// Wave: 16 pixels x all 64 out channels (4 C-tiles). Block = 8 waves = 128 px.
// Safe in-place (x == y): wave reads only its own 16 pixels before storing.
// NPIX is a compile-time constant so channel strides fold into immediates
// instead of per-load v_mul_u64 address math.
// ---------------------------------------------------------------------------
template <int NPIX>
__global__ __launch_bounds__(256) void conv1x1_gn_kernel(
    const float* __restrict__ x, const float* __restrict__ w,
    const float* __restrict__ bias, float* __restrict__ y,
    float* __restrict__ stats) {
  __shared__ float sSum[16];
  __shared__ float sSsq[16];
  if (threadIdx.x < 16) { sSum[threadIdx.x] = 0.0f; sSsq[threadIdx.x] = 0.0f; }
  __syncthreads();

  const int b    = blockIdx.y;
  const int wave = threadIdx.x >> 5;
  const int lane = threadIdx.x & 31;
  const int ksel = lane >> 4;       // half-wave selects K pair
  const int n    = lane & 15;       // pixel / out-row index within tile
  const int p0   = (blockIdx.x * 8 + wave) * 16;

  const float* xb = x + (size_t)b * Cn * NPIX + p0 + n + (size_t)(2 * ksel) * NPIX;
  v8f acc[4] = {};
  #pragma unroll
  for (int k0 = 0; k0 < 64; k0 += 4) {
    v2f bf;
    bf.x = xb[(size_t)k0 * NPIX];
    bf.y = xb[(size_t)(k0 + 1) * NPIX];
    const int kk = k0 + 2 * ksel;
    #pragma unroll
    for (int mt = 0; mt < 4; ++mt) {
      v2f a;
      a.x = w[(mt * 16 + n) * 64 + kk];
      a.y = w[(mt * 16 + n) * 64 + kk + 1];
      acc[mt] = wmma4(a, bf, acc[mt]);
    }
  }

  float* yb = y + (size_t)b * Cn * NPIX + p0 + n;
  float lsum[8] = {0,0,0,0,0,0,0,0};
  float lssq[8] = {0,0,0,0,0,0,0,0};
  #pragma unroll
  for (int mt = 0; mt < 4; ++mt) {
    #pragma unroll
    for (int r = 0; r < 8; ++r) {
      const int o = mt * 16 + r + 8 * ksel;
      float v = acc[mt][r] + bias[o];
      yb[(size_t)o * NPIX] = v;
      const int slot = mt * 2 + (r >> 2);
      lsum[slot] += v;
      lssq[slot] += v * v;
    }
  }
  #pragma unroll
  for (int slot = 0; slot < 8; ++slot) {
    const int g = (slot >> 1) * 4 + 2 * ksel + (slot & 1);
    atomicAdd(&sSum[g], lsum[slot]);
    atomicAdd(&sSsq[g], lssq[slot]);
  }
  __syncthreads();
  if (threadIdx.x < 16) {
    atomicAdd(&stats[b * 16 + threadIdx.x], sSum[threadIdx.x]);
    atomicAdd(&stats[32 + b * 16 + threadIdx.x], sSsq[threadIdx.x]);
  }
}

// stats -> per (b,c) scale/shift
__global__ void gn_finalize_kernel(float* stats, const float* __restrict__ gamma,
                                   const float* __restrict__ beta, int npix) {
  int i = threadIdx.x;                 // 0..127 = b*64+c
  if (i >= Bn * Cn) return;
  int b = i >> 6, c = i & 63, g = c >> 2;
  float N = 4.0f * (float)npix;
  float mu  = stats[b * 16 + g] / N;
  float var = stats[32 + b * 16 + g] / N - mu * mu;
  float rs  = rsqrtf(fmaxf(var, 0.0f) + 1e-5f);
  float sc  = gamma[c] * rs;
  stats[64  + i] = sc;
  stats[192 + i] = beta[c] - mu * sc;
}

template <int SHIFT>
__global__ __launch_bounds__(256) void gn_apply_kernel(float* y,
                                                       const float* __restrict__ stats) {
  size_t idx = (size_t)blockIdx.x * 256 + threadIdx.x;
  int bc = (int)(idx >> SHIFT);        // b*64+c
  y[idx] = y[idx] * stats[64 + bc] + stats[192 + bc];
}

// ---------------------------------------------------------------------------
// score[b,h,wq,v] = sum_c base_fea[b,c,h,wq] * ref_fea[b,c,h,v]
// Per (b,h): GEMM M=wq(256) N=v(256) K=c(64). Wave = 16m x 64n block.
// grid (8, Bn*256), block 256.
// ---------------------------------------------------------------------------
__global__ __launch_bounds__(256) void attn_score_kernel(
    const float* __restrict__ bfea, const float* __restrict__ rfea,
    float* __restrict__ S) {
  const int bh = blockIdx.y;
  const int b = bh >> 8, h = bh & 255;
  const int wave = threadIdx.x >> 5, lane = threadIdx.x & 31;
  const int ksel = lane >> 4, n = lane & 15;
  const int blk = blockIdx.x * 8 + wave;   // 0..63
  const int mt = blk >> 2;                 // 0..15
  const int n0 = (blk & 3) * 64;

  const size_t base = (size_t)b * Cn * HWP + (size_t)h * Wp + (size_t)(2 * ksel) * HWP;
  const float* A  = bfea + base + mt * 16 + n;  // A[k][m]
  const float* Bm = rfea + base + n0 + n;

  v8f acc[4] = {};
  #pragma unroll
  for (int k0 = 0; k0 < 64; k0 += 4) {
    v2f a;
    a.x = A[(size_t)k0 * HWP];
    a.y = A[(size_t)(k0 + 1) * HWP];
    #pragma unroll
    for (int nt = 0; nt < 4; ++nt) {
      v2f bf;
      bf.x = Bm[(size_t)k0 * HWP + nt * 16];
      bf.y = Bm[(size_t)(k0 + 1) * HWP + nt * 16];
      acc[nt] = wmma4(a, bf, acc[nt]);
    }
  }
  float* Sout = S + (size_t)bh * 65536;   // [wq][v]
  #pragma unroll
  for (int nt = 0; nt < 4; ++nt)
    #pragma unroll
    for (int r = 0; r < 8; ++r)
      Sout[(size_t)(mt * 16 + r + 8 * ksel) * 256 + n0 + nt * 16 + n] = acc[nt][r];
}

// softmax over last axis (256) of S; one wave per row.
__global__ __launch_bounds__(256) void softmax_kernel(float* S) {
  const int wave = threadIdx.x >> 5, lane = threadIdx.x & 31;
  const int row = blockIdx.x * 8 + wave;       // Bn*256*256 rows
  float* p = S + (size_t)row * 256;
  float v[8];
  #pragma unroll
  for (int j = 0; j < 8; ++j) v[j] = p[lane + 32 * j];
  float mx = v[0];
  #pragma unroll
  for (int j = 1; j < 8; ++j) mx = fmaxf(mx, v[j]);
  #pragma unroll
  for (int off = 16; off > 0; off >>= 1) mx = fmaxf(mx, __shfl_xor(mx, off, 32));
  float s = 0.0f;
  #pragma unroll
  for (int j = 0; j < 8; ++j) { v[j] = __expf(v[j] - mx); s += v[j]; }
  #pragma unroll
  for (int off = 16; off > 0; off >>= 1) s += __shfl_xor(s, off, 32);
  float inv = 1.0f / s;
  #pragma unroll
  for (int j = 0; j < 8; ++j) p[lane + 32 * j] = v[j] * inv;
}

// warp[b,c,h,wq] = sum_v M[b,h,wq,v] * ref_info[b,c,h,v]
// Per (b,h): GEMM M=c(64) N=wq(256) K=v(256). Wave = 16m x 64n. grid (2, Bn*256).
__global__ __launch_bounds__(256) void attn_warp_kernel(
    const float* __restrict__ S, const float* __restrict__ rinfo,
    float* __restrict__ wrp) {
  const int bh = blockIdx.y;
  const int b = bh >> 8, h = bh & 255;
  const int wave = threadIdx.x >> 5, lane = threadIdx.x & 31;
  const int ksel = lane >> 4, n = lane & 15;
  const int blk = blockIdx.x * 8 + wave;   // 0..15
  const int mt = blk >> 2;                 // c tile 0..3
  const int n0 = (blk & 3) * 64;

  const float* A  = rinfo + (size_t)b * Cn * HWP + (size_t)h * Wp
                    + (size_t)(mt * 16 + n) * HWP + 2 * ksel;     // A[m][k]
  const float* Bm = S + (size_t)bh * 65536
                    + (size_t)(n0 + n) * 256 + 2 * ksel;          // B[k][n] = S[n][k]

  v8f acc[4] = {};
  for (int k0 = 0; k0 < 256; k0 += 4) {
    v2f a;
    a.x = A[k0];
    a.y = A[k0 + 1];
    #pragma unroll
    for (int nt = 0; nt < 4; ++nt) {
      v2f bf;
      bf.x = Bm[(size_t)(nt * 16) * 256 + k0];
      bf.y = Bm[(size_t)(nt * 16) * 256 + k0 + 1];
      acc[nt] = wmma4(a, bf, acc[nt]);
    }
  }
  float* W = wrp + (size_t)b * Cn * HWP + (size_t)h * Wp;
  #pragma unroll
  for (int nt = 0; nt < 4; ++nt)
    #pragma unroll
    for (int r = 0; r < 8; ++r)
      W[(size_t)(mt * 16 + r + 8 * ksel) * HWP + n0 + nt * 16 + n] = acc[nt][r];
}

// ---------------------------------------------------------------------------
// bicubic x2, align_corners=True, a=-0.75, border-clamped taps.
// ---------------------------------------------------------------------------
__device__ __forceinline__ void cubw(float t, float w[4]) {
  const float A = -0.75f;
  float x0 = t + 1.0f;
  w[0] = ((A * x0 - 5.0f * A) * x0 + 8.0f * A) * x0 - 4.0f * A;
  w[1] = ((A + 2.0f) * t - (A + 3.0f)) * t * t + 1.0f;
  float u = 1.0f - t;
  w[2] = ((A + 2.0f) * u - (A + 3.0f)) * u * u + 1.0f;
  float x3 = 2.0f - t;
  w[3] = ((A * x3 - 5.0f * A) * x3 + 8.0f * A) * x3 - 4.0f * A;
}

__global__ __launch_bounds__(256) void bicubic_up2_kernel(const float* __restrict__ in,
                                                          float* __restrict__ out) {
  size_t idx = (size_t)blockIdx.x * 256 + threadIdx.x;  // Bn*Cn*HWF
  int X = (int)(idx & 511), Y = (int)((idx >> 9) & 511);
  int bc = (int)(idx >> 18);
  float sy = (float)Y * (255.0f / 511.0f);
  float sx = (float)X * (255.0f / 511.0f);
  int iy = (int)floorf(sy), ix = (int)floorf(sx);
  float ty = sy - (float)iy, tx = sx - (float)ix;
  float wy[4], wx[4];
  cubw(ty, wy); cubw(tx, wx);
  const float* src = in + (size_t)bc * HWP;
  float acc = 0.0f;
  #pragma unroll
  for (int dy = 0; dy < 4; ++dy) {
    int yy = min(max(iy - 1 + dy, 0), 255);
    const float* row = src + (size_t)yy * Wp;
    float rs = 0.0f;
    #pragma unroll
    for (int dx = 0; dx < 4; ++dx) {
      int xx = min(max(ix - 1 + dx, 0), 255);
      rs += wx[dx] * row[xx];
    }
    acc += wy[dy] * rs;
  }
  out[idx] = acc;
}

// ---------------------------------------------------------------------------
// conv3x3 (implicit GEMM, WMMA). K ordered (ky,kx) outer, channel inner.
// Wave: 64 out-ch x 64 pixels (one row segment). Block = 8 waves = one 512 row.
// ---------------------------------------------------------------------------
__device__ __forceinline__ const float* chanptr(const float* f, const float* bi, int c) {
  return (c < 64) ? (f + (size_t)c * HWF) : (bi + (size_t)(c - 64) * HWF);
}

__global__ __launch_bounds__(256) void conv3x3_r1_kernel(
    const float* __restrict__ fused, const float* __restrict__ binfo,
    const float* __restrict__ w, const float* __restrict__ bias,
    float* __restrict__ y1) {
  const int b = blockIdx.y;
  const int yrow = blockIdx.x;
  const int wave = threadIdx.x >> 5, lane = threadIdx.x & 31;
  const int ksel = lane >> 4, n = lane & 15;
  const int x0 = wave * 64;
  const float* fb = fused + (size_t)b * Cn * HWF;
  const float* bb = binfo + (size_t)b * Cn * HWF;

  v8f acc[4][4] = {};
  for (int t9 = 0; t9 < 9; ++t9) {
    const int ky = t9 / 3, kx = t9 - ky * 3;
    const int yy = yrow + ky - 1;
    const bool yok = (unsigned)yy < (unsigned)Hf;
    const size_t rowoff = (size_t)yy * Wf;
    for (int i0 = 0; i0 < 128; i0 += 4) {
      const int kk = i0 + 2 * ksel;
      v2f a[4];
      #pragma unroll
      for (int mt = 0; mt < 4; ++mt) {
        const int o = mt * 16 + n;
        a[mt].x = w[(size_t)o * 1152 + kk * 9 + t9];
        a[mt].y = w[(size_t)o * 1152 + (kk + 1) * 9 + t9];
      }
      const float* s0 = chanptr(fb, bb, kk);
      const float* s1 = chanptr(fb, bb, kk + 1);
      v2f bf[4];
      #pragma unroll
      for (int nt = 0; nt < 4; ++nt) {
        const int xx = x0 + nt * 16 + n + kx - 1;
        const bool ok = yok && ((unsigned)xx < (unsigned)Wf);
        const size_t poff = rowoff + xx;
        bf[nt].x = ok ? s0[poff] : 0.0f;
        bf[nt].y = ok ? s1[poff] : 0.0f;
      }
      #pragma unroll
      for (int mt = 0; mt < 4; ++mt)
        #pragma unroll
        for (int nt = 0; nt < 4; ++nt)
          acc[mt][nt] = wmma4(a[mt], bf[nt], acc[mt][nt]);
    }
  }
  float* out = y1 + (size_t)b * Cn * HWF + (size_t)yrow * Wf;
  #pragma unroll
  for (int mt = 0; mt < 4; ++mt)
    #pragma unroll
    for (int nt = 0; nt < 4; ++nt)
      #pragma unroll
      for (int r = 0; r < 8; ++r) {
        const int o = mt * 16 + r + 8 * ksel;
        float v = acc[mt][nt][r] + bias[o];
        out[(size_t)o * HWF + x0 + nt * 16 + n] = fmaxf(v, 0.0f);
      }
}

__global__ __launch_bounds__(256) void conv3x3_r2_skip_kernel(
    const float* __restrict__ y1, const float* __restrict__ fused,
    const float* __restrict__ binfo, const float* __restrict__ w2,
    const float* __restrict__ b2, const float* __restrict__ wrs,
    const float* __restrict__ brs, float* __restrict__ out) {
  const int b = blockIdx.y;
  const int yrow = blockIdx.x;
  const int wave = threadIdx.x >> 5, lane = threadIdx.x & 31;
  const int ksel = lane >> 4, n = lane & 15;
  const int x0 = wave * 64;
  const float* yb = y1 + (size_t)b * Cn * HWF;
  const float* fb = fused + (size_t)b * Cn * HWF;
  const float* bb = binfo + (size_t)b * Cn * HWF;

  v8f acc[4][4] = {};
  // conv3x3 over y1 (64 in-ch)
  for (int t9 = 0; t9 < 9; ++t9) {
    const int ky = t9 / 3, kx = t9 - ky * 3;
    const int yy = yrow + ky - 1;
    const bool yok = (unsigned)yy < (unsigned)Hf;
    const size_t rowoff = (size_t)yy * Wf;
    for (int i0 = 0; i0 < 64; i0 += 4) {
      const int kk = i0 + 2 * ksel;
      v2f a[4];
      #pragma unroll
      for (int mt = 0; mt < 4; ++mt) {
        const int o = mt * 16 + n;
        a[mt].x = w2[(size_t)o * 576 + kk * 9 + t9];
        a[mt].y = w2[(size_t)o * 576 + (kk + 1) * 9 + t9];
      }
      const float* s0 = yb + (size_t)kk * HWF;
      const float* s1 = yb + (size_t)(kk + 1) * HWF;
      v2f bf[4];
      #pragma unroll
      for (int nt = 0; nt < 4; ++nt) {
        const int xx = x0 + nt * 16 + n + kx - 1;
        const bool ok = yok && ((unsigned)xx < (unsigned)Wf);
        const size_t poff = rowoff + xx;
        bf[nt].x = ok ? s0[poff] : 0.0f;
        bf[nt].y = ok ? s1[poff] : 0.0f;
      }
      #pragma unroll
      for (int mt = 0; mt < 4; ++mt)
        #pragma unroll
        for (int nt = 0; nt < 4; ++nt)
          acc[mt][nt] = wmma4(a[mt], bf[nt], acc[mt][nt]);
    }
  }
  // fused skip projection: wrs[64x128] @ concat(fused, binfo) at center pixel
  const size_t crow = (size_t)yrow * Wf;
  for (int i0 = 0; i0 < 128; i0 += 4) {
    const int kk = i0 + 2 * ksel;
    v2f a[4];
    #pragma unroll
    for (int mt = 0; mt < 4; ++mt) {
      const int o = mt * 16 + n;
      a[mt].x = wrs[o * 128 + kk];
      a[mt].y = wrs[o * 128 + kk + 1];
    }
    const float* s0 = chanptr(fb, bb, kk);
    const float* s1 = chanptr(fb, bb, kk + 1);
    v2f bf[4];
    #pragma unroll
    for (int nt = 0; nt < 4; ++nt) {
      const size_t poff = crow + x0 + nt * 16 + n;
      bf[nt].x = s0[poff];
      bf[nt].y = s1[poff];
    }
    #pragma unroll
    for (int mt = 0; mt < 4; ++mt)
      #pragma unroll
      for (int nt = 0; nt < 4; ++nt)
        acc[mt][nt] = wmma4(a[mt], bf[nt], acc[mt][nt]);
  }
  float* op = out + (size_t)b * Cn * HWF + (size_t)yrow * Wf;
  #pragma unroll
  for (int mt = 0; mt < 4; ++mt)
    #pragma unroll
    for (int nt = 0; nt < 4; ++nt)
      #pragma unroll
      for (int r = 0; r < 8; ++r) {
        const int o = mt * 16 + r + 8 * ksel;
        op[(size_t)o * HWF + x0 + nt * 16 + n] = acc[mt][nt][r] + b2[o] + brs[o];
      }
}

// ---------------------------------------------------------------------------
// Orchestration
// ---------------------------------------------------------------------------
extern "C" void kernel_launch(void* const* d_in, const int* in_sizes, int n_in,
                              void* d_out, int out_size, void* d_ws, size_t ws_size,
                              hipStream_t stream) {
  const float* base    = (const float*)d_in[0];
  const float* ref     = (const float*)d_in[1];
  const float* w_fea1  = (const float*)d_in[2];
  const float* b_fea1  = (const float*)d_in[3];
  const float* g_fea1  = (const float*)d_in[4];
  const float* be_fea1 = (const float*)d_in[5];
  const float* w_fea2  = (const float*)d_in[6];
  const float* b_fea2  = (const float*)d_in[7];
  const float* g_fea2  = (const float*)d_in[8];
  const float* be_fea2 = (const float*)d_in[9];
  const float* w_info1 = (const float*)d_in[10];
  const float* b_info1 = (const float*)d_in[11];
  const float* g_info1 = (const float*)d_in[12];
  const float* be_info1= (const float*)d_in[13];
  const float* w_info2 = (const float*)d_in[14];
  const float* b_info2 = (const float*)d_in[15];
  const float* g_info2 = (const float*)d_in[16];
  const float* be_info2= (const float*)d_in[17];
  const float* w_fus   = (const float*)d_in[18];
  const float* b_fus   = (const float*)d_in[19];
  const float* g_fus   = (const float*)d_in[20];
  const float* be_fus  = (const float*)d_in[21];
  const float* w_r1    = (const float*)d_in[22];
  const float* b_r1    = (const float*)d_in[23];
  const float* w_r2    = (const float*)d_in[24];
  const float* b_r2    = (const float*)d_in[25];
  const float* w_rs    = (const float*)d_in[26];
  const float* b_rs    = (const float*)d_in[27];

  float* ws = (float*)d_ws;
  const size_t BIGE = (size_t)Bn * Cn * HWF;      // 33,554,432 floats
  const size_t PE   = (size_t)Bn * Cn * HWP;      //  8,388,608 floats
  float* BIG0  = ws;                   // ref_info (head) -> base_info (full)
  float* BIG1  = ws + BIGE;            // score S -> up -> fused (in place)
  float* BIG2  = ws + 2 * BIGE;        // {P0,P1,BFEA,RFEA} -> warp -> y1
  float* stats = ws + 3 * BIGE;        // 320 floats
  float* P0    = BIG2;
  float* P1    = BIG2 + PE;
  float* BFEA  = BIG2 + 2 * PE;
  float* RFEA  = BIG2 + 3 * PE;
  float* RINFO = BIG0;
  float* WARP  = BIG2;
  float* out   = (float*)d_out;

  dim3 blk(256);

  // 1) maxpool both inputs
  maxpool_kernel<<<dim3((unsigned)(PE / 256)), blk, 0, stream>>>(base, P0);
  maxpool_kernel<<<dim3((unsigned)(PE / 256)), blk, 0, stream>>>(ref, P1);

  auto convgn_p = [&](const float* x, const float* w, const float* bias,
                      const float* gamma, const float* beta, float* y) {
    zero_stats_kernel<<<1, 64, 0, stream>>>(stats);
    conv1x1_gn_kernel<HWP><<<dim3(HWP / 128, Bn), blk, 0, stream>>>(x, w, bias, y, stats);
    gn_finalize_kernel<<<1, 128, 0, stream>>>(stats, gamma, beta, HWP);
    gn_apply_kernel<16><<<dim3((unsigned)(PE / 256)), blk, 0, stream>>>(y, stats);
  };
  auto convgn_f = [&](const float* x, const float* w, const float* bias,
                      const float* gamma, const float* beta, float* y) {
    zero_stats_kernel<<<1, 64, 0, stream>>>(stats);
    conv1x1_gn_kernel<HWF><<<dim3(HWF / 128, Bn), blk, 0, stream>>>(x, w, bias, y, stats);
    gn_finalize_kernel<<<1, 128, 0, stream>>>(stats, gamma, beta, HWF);
    gn_apply_kernel<18><<<dim3((unsigned)(BIGE / 256)), blk, 0, stream>>>(y, stats);
  };

  // 2) pooled 1x1 + GN
  convgn_p(P1, w_info1, b_info1, g_info1, be_info1, RINFO);
  convgn_p(P1, w_fea1,  b_fea1,  g_fea1,  be_fea1,  RFEA);
  convgn_p(P0, w_fea2,  b_fea2,  g_fea2,  be_fea2,  BFEA);

  // 3) row-wise attention
  attn_score_kernel<<<dim3(8, Bn * 256), blk, 0, stream>>>(BFEA, RFEA, BIG1);
  softmax_kernel<<<dim3((unsigned)(Bn * 256 * 256 / 8)), blk, 0, stream>>>(BIG1);
  attn_warp_kernel<<<dim3(2, Bn * 256), blk, 0, stream>>>(BIG1, RINFO, WARP);

  // 4) bicubic x2 upsample: warp -> up (BIG1, overwrites dead S)
  bicubic_up2_kernel<<<dim3((unsigned)(BIGE / 256)), blk, 0, stream>>>(WARP, BIG1);

  // 5) base_info (deferred so it can overwrite dead ref_info in BIG0)
  convgn_f(base, w_info2, b_info2, g_info2, be_info2, BIG0);

  // 6) fused = conv1x1_gn(up) in place in BIG1
  convgn_f(BIG1, w_fus, b_fus, g_fus, be_fus, BIG1);

  // 7) ResBlock: conv3x3(2C->C)+relu, then conv3x3(C->C) + 1x1 skip(2C->C)
  conv3x3_r1_kernel<<<dim3(Hf, Bn), blk, 0, stream>>>(BIG1, BIG0, w_r1, b_r1, BIG2);
  conv3x3_r2_skip_kernel<<<dim3(Hf, Bn), blk, 0, stream>>>(BIG2, BIG1, BIG0, w_r2, b_r2,
                                                           w_rs, b_rs, out);
}